// GraphSage_7919919693881
// MI455X (gfx1250) — compile-verified
//
#include <hip/hip_runtime.h>

typedef __attribute__((ext_vector_type(2))) float v2f;
typedef __attribute__((ext_vector_type(8))) float v8f;

// ---------------------------------------------------------------- utilities
__global__ void zero_f4_kernel(float4* __restrict__ p, int n4) {
    int i = blockIdx.x * 256 + threadIdx.x;
    if (i < n4) p[i] = make_float4(0.f, 0.f, 0.f, 0.f);
}

// in-degree accumulation (float so it feeds 1/max(deg,1) directly)
__global__ void degree_kernel(const int* __restrict__ dst, float* __restrict__ deg, int nE) {
    int e = blockIdx.x * 256 + threadIdx.x;
    if (e < nE) atomicAdd(&deg[dst[e]], 1.0f);
}

__global__ void invdeg_kernel(float* __restrict__ deg, int n) {
    int i = blockIdx.x * 256 + threadIdx.x;
    if (i < n) deg[i] = 1.0f / fmaxf(deg[i], 1.0f);
}

// ---------------------------------------------------------------- scatter
// One wave per edge: 32 lanes * float4 = one coalesced 512B row of h[src],
// 128 f32 atomic adds into agg[dst] (resolve in L2; working set is L2-resident).
__global__ void scatter_kernel(const float* __restrict__ h,
                               const int* __restrict__ src,
                               const int* __restrict__ dst,
                               float* __restrict__ agg, int nE) {
    int tid = blockIdx.x * 256 + threadIdx.x;
    int e = tid >> 5;
    if (e >= nE) return;
    int f = (tid & 31) << 2;                  // lane*4 features
    int s = src[e];
    int d = dst[e];
    float4 v = *(const float4*)&h[(size_t)s * 128 + f];
    float* a = &agg[(size_t)d * 128 + f];
    atomicAdd(a + 0, v.x);
    atomicAdd(a + 1, v.y);
    atomicAdd(a + 2, v.z);
    atomicAdd(a + 3, v.w);
}

// ---------------------------------------------------------------- fused SAGE GEMM
// Out[m, :] = act( Hin[m,:] @ Ws + (Agg[m,:] * invdeg[m]) @ Wn + bias )
// Block: 256 threads (8 waves), 64-row stripe staged in LDS.
// Each wave owns ONE 16-column tile and NRT row tiles; B (weight) fragments
// are loaded once per k-step and reused across all NRT accumulators.
// K = 128, v_wmma_f32_16x16x4_f32.
template <int OUT, bool RELU>
__global__ __launch_bounds__(256) void sage_gemm_kernel(
    const float* __restrict__ Hin, const float* __restrict__ Agg,
    const float* __restrict__ invdeg,
    const float* __restrict__ Ws, const float* __restrict__ Wn,
    const float* __restrict__ bias,
    float* __restrict__ Out, int n) {
    __shared__ float sH[64 * 128];
    __shared__ float sG[64 * 128];

    const int tid = threadIdx.x;
    const int rowBase = blockIdx.x * 64;
    const bool full = (rowBase + 64 <= n);   // block-uniform fast path

    // ---- stage 64 rows of Hin and deg-scaled Agg into LDS (float4, coalesced)
    if (full) {
        for (int i = tid; i < 64 * 32; i += 256) {
            int gr = rowBase + (i >> 5);
            int c4 = i & 31;
            float s = invdeg[gr];
            float4 h4 = ((const float4*)Hin)[(size_t)gr * 32 + c4];
            float4 t  = ((const float4*)Agg)[(size_t)gr * 32 + c4];
            ((float4*)sH)[i] = h4;
            ((float4*)sG)[i] = make_float4(t.x * s, t.y * s, t.z * s, t.w * s);
        }
    } else {
        for (int i = tid; i < 64 * 32; i += 256) {
            int gr = rowBase + (i >> 5);
            int c4 = i & 31;
            float4 h4 = make_float4(0.f, 0.f, 0.f, 0.f);
            float4 g4 = h4;
            if (gr < n) {
                h4 = ((const float4*)Hin)[(size_t)gr * 32 + c4];
                float4 t = ((const float4*)Agg)[(size_t)gr * 32 + c4];
                float s = invdeg[gr];
                g4 = make_float4(t.x * s, t.y * s, t.z * s, t.w * s);
            }
            ((float4*)sH)[i] = h4;
            ((float4*)sG)[i] = g4;
        }
    }
    __syncthreads();

    const int wave = tid >> 5;
    const int lane = tid & 31;
    const int hf   = lane >> 4;   // half-wave: K base +2, M base +8
    const int lh   = lane & 15;

    constexpr int CT   = OUT / 16;      // column tiles (8 or 4)
    constexpr int NRT  = (4 * CT) / 8;  // row tiles per wave (4 or 2)
    constexpr int RSTR = 8 / CT;        // row-tile stride (1 or 2)

    const int co  = (wave % CT) * 16;   // this wave's (only) column tile
    const int rt0 = wave / CT;

    // accumulators, init = bias (column-dependent only; same for all row tiles)
    v8f c[NRT];
    const float bv = bias[co + lh];
#pragma unroll
    for (int i = 0; i < NRT; ++i)
#pragma unroll
        for (int j = 0; j < 8; ++j) c[i][j] = bv;

    int aoff[NRT];
#pragma unroll
    for (int i = 0; i < NRT; ++i)
        aoff[i] = ((rt0 + i * RSTR) * 16 + lh) * 128;

    const float* wsp = Ws + co + lh;    // B fragment base (per-lane column)
    const float* wnp = Wn + co + lh;

#pragma unroll 2
    for (int kk = 0; kk < 128; kk += 4) {
        const int k0 = kk + 2 * hf;
        // B fragments: loaded ONCE per k-step, reused by all NRT row tiles
        v2f bs, bn;
        bs.x = wsp[(k0 + 0) * OUT];
        bs.y = wsp[(k0 + 1) * OUT];
        bn.x = wnp[(k0 + 0) * OUT];
        bn.y = wnp[(k0 + 1) * OUT];

        v2f ah[NRT], ag[NRT];
#pragma unroll
        for (int i = 0; i < NRT; ++i) {
            ah[i] = *(const v2f*)&sH[aoff[i] + k0];
            ag[i] = *(const v2f*)&sG[aoff[i] + k0];
        }
        // self path (shared B = Ws tile)
#pragma unroll
        for (int i = 0; i < NRT; ++i)
            c[i] = __builtin_amdgcn_wmma_f32_16x16x4_f32(false, ah[i], false, bs,
                                                         (short)0, c[i], false, false);
        // neighbor path (shared B = Wn tile)
#pragma unroll
        for (int i = 0; i < NRT; ++i)
            c[i] = __builtin_amdgcn_wmma_f32_16x16x4_f32(false, ag[i], false, bn,
                                                         (short)0, c[i], false, false);
    }

    // C/D layout: VGPR j -> row j (lanes 0-15) / row j+8 (lanes 16-31), col = lh
#pragma unroll
    for (int i = 0; i < NRT; ++i) {
        const int gr0 = rowBase + (rt0 + i * RSTR) * 16 + hf * 8;
        if (full) {
            float* op = Out + (size_t)gr0 * OUT + co + lh;
#pragma unroll
            for (int j = 0; j < 8; ++j) {
                float v = c[i][j];
                if (RELU) v = fmaxf(v, 0.f);
                op[(size_t)j * OUT] = v;
            }
        } else {
#pragma unroll
            for (int j = 0; j < 8; ++j) {
                int gr = gr0 + j;
                if (gr < n) {
                    float v = c[i][j];
                    if (RELU) v = fmaxf(v, 0.f);
                    Out[(size_t)gr * OUT + co + lh] = v;
                }
            }
        }
    }
}

// ---------------------------------------------------------------- launch
extern "C" void kernel_launch(void* const* d_in, const int* in_sizes, int n_in,
                              void* d_out, int out_size, void* d_ws, size_t ws_size,
                              hipStream_t stream) {
    (void)n_in; (void)out_size; (void)ws_size;

    const float* x   = (const float*)d_in[0];
    const int*   src = (const int*)d_in[1];
    const int*   dst = (const int*)d_in[2];
    const float* Ws0 = (const float*)d_in[3];
    const float* Wn0 = (const float*)d_in[4];
    const float* b0  = (const float*)d_in[5];
    const float* Ws1 = (const float*)d_in[6];
    const float* Wn1 = (const float*)d_in[7];
    const float* b1  = (const float*)d_in[8];
    const float* Ws2 = (const float*)d_in[9];
    const float* Wn2 = (const float*)d_in[10];
    const float* b2  = (const float*)d_in[11];
    float* out = (float*)d_out;

    const int N = in_sizes[0] / 128;  // 100000 nodes
    const int E = in_sizes[1];        // 1600000 edges

    // workspace layout (floats): [deg/invdeg | agg (N*128) | h (N*128)] ~ 103 MB
    float* deg = (float*)d_ws;
    float* agg = deg + 102400;                  // 16B-aligned offsets
    float* h   = agg + (size_t)N * 128;

    const int nf     = N * 128;
    const int n4     = nf / 4;
    const int zBlks  = (n4 + 255) / 256;
    const int dZBlks = (N / 4 + 255) / 256;
    const int eBlks  = (E + 255) / 256;
    const int nBlks  = (N + 255) / 256;
    const int sBlks  = (int)(((long long)E * 32 + 255) / 256);
    const int gBlks  = (N + 63) / 64;

    // degrees -> 1/max(deg,1)
    zero_f4_kernel<<<dZBlks, 256, 0, stream>>>((float4*)deg, N / 4);
    degree_kernel<<<eBlks, 256, 0, stream>>>(dst, deg, E);
    invdeg_kernel<<<nBlks, 256, 0, stream>>>(deg, N);

    // layer 0: x -> h (relu)
    zero_f4_kernel<<<zBlks, 256, 0, stream>>>((float4*)agg, n4);
    scatter_kernel<<<sBlks, 256, 0, stream>>>(x, src, dst, agg, E);
    sage_gemm_kernel<128, true><<<gBlks, 256, 0, stream>>>(x, agg, deg, Ws0, Wn0, b0, h, N);

    // layer 1: h -> h (in place; each block reads/writes only its own rows)
    zero_f4_kernel<<<zBlks, 256, 0, stream>>>((float4*)agg, n4);
    scatter_kernel<<<sBlks, 256, 0, stream>>>(h, src, dst, agg, E);
    sage_gemm_kernel<128, true><<<gBlks, 256, 0, stream>>>(h, agg, deg, Ws1, Wn1, b1, h, N);

    // layer 2: h -> out (64 classes, no relu)
    zero_f4_kernel<<<zBlks, 256, 0, stream>>>((float4*)agg, n4);
    scatter_kernel<<<sBlks, 256, 0, stream>>>(h, src, dst, agg, E);
    sage_gemm_kernel<64, false><<<gBlks, 256, 0, stream>>>(h, agg, deg, Ws2, Wn2, b2, out, N);
}